// UpdateNetwork_54674933678726
// MI455X (gfx1250) — compile-verified
//
#include <hip/hip_runtime.h>

// ---------------------------------------------------------------------------
// Logic-gate network, gfx1250 (MI455X).
//   Kernel 1: c = softmax(W) @ GATE_COEFFS via chained V_WMMA_F32_16X16X4_F32.
//   Kernel 2: 23-layer gather/eval sweep, LDS-resident activations,
//             float2 (b64) LDS traffic, scalarized per-node metadata.
// ---------------------------------------------------------------------------

typedef __attribute__((ext_vector_type(2))) float v2f;
typedef __attribute__((ext_vector_type(8))) float v8f;

#define TOTAL_NODES 2175
#define N_CELLS     262144
#define N_LAYERS    23

#define CELLS    128      // cells per workgroup tile
#define LSTRIDE  132      // LDS row stride in floats (132 % 64 == 4 -> transpose conflict-free)
#define THREADS  256

// ---------------------------------------------------------------------------
// Per-lane B-operand table for the 16x16x4 f32 WMMA, built at compile time.
// B is the 4x16 chunk of GATE_COEFFS (rows 4k..4k+3) padding N to 16:
//   lane layout: lanes 0-15: VGPR0=K0, VGPR1=K1 ; lanes 16-31: VGPR0=K2, VGPR1=K3
//   N = lane & 15 (only N<4 nonzero).
// t.v[lane][2k+0/1] = B chunk k, this lane's two K rows.
// ---------------------------------------------------------------------------
struct BTab { float v[32][8]; };

static constexpr BTab makeBTab() {
    constexpr float G[16][4] = {
        {0,0,0,0},{0,0,0,1},{0,1,0,-1},{0,1,0,0},
        {0,0,1,-1},{0,0,1,0},{0,1,1,-2},{0,1,1,-1},
        {1,-1,-1,1},{1,-1,-1,2},{1,0,-1,0},{1,0,-1,1},
        {1,-1,0,0},{1,-1,0,1},{1,0,0,-1},{1,0,0,0}};
    BTab t{};
    for (int lane = 0; lane < 32; ++lane) {
        const int khalf = (lane < 16) ? 0 : 2;
        const int col   = lane & 15;
        for (int k = 0; k < 4; ++k) {
            t.v[lane][2 * k + 0] = (col < 4) ? G[4 * k + khalf    ][col] : 0.0f;
            t.v[lane][2 * k + 1] = (col < 4) ? G[4 * k + khalf + 1][col] : 0.0f;
        }
    }
    return t;
}
__device__ const BTab kBTab = makeBTab();

// ---------------------------------------------------------------------------
// Kernel 1: one wave32 per 16-node tile.
//   p(16x16) @ G(16x4 padded) = c(16x4), via 4 chained 16x16x4 f32 WMMAs.
// ---------------------------------------------------------------------------
__global__ __launch_bounds__(32)
void logicgate_coeff_kernel(const float* __restrict__ W, float* __restrict__ coeff)
{
    const int  lane = threadIdx.x;        // 0..31
    const int  row  = lane & 15;          // M (node within tile)
    const bool hi   = lane >= 16;         // K-half selector
    const int  tile = blockIdx.x;

    const int node  = tile * 16 + row;
    const int nodeC = node < TOTAL_NODES ? node : (TOTAL_NODES - 1);

    // Load the 16 logits as 4x float4 (b128, coalesced across lanes).
    const float4* __restrict__ W4 = (const float4*)W;
    float w[16];
    #pragma unroll
    for (int q = 0; q < 4; ++q) {
        const float4 t = W4[nodeC * 4 + q];
        w[4 * q + 0] = t.x; w[4 * q + 1] = t.y; w[4 * q + 2] = t.z; w[4 * q + 3] = t.w;
    }

    // softmax (temperature = 1)
    float m = w[0];
    #pragma unroll
    for (int k = 1; k < 16; ++k) m = fmaxf(m, w[k]);
    float p[16];
    float sum = 0.0f;
    #pragma unroll
    for (int k = 0; k < 16; ++k) { p[k] = __expf(w[k] - m); sum += p[k]; }
    const float inv = __builtin_amdgcn_rcpf(sum);   // v_rcp_f32

    const float* __restrict__ bl = kBTab.v[lane];
    v8f acc = {};
    #pragma unroll
    for (int k = 0; k < 4; ++k) {        // K split: gates 4k .. 4k+3
        // A 16x4 layout: lanes 0-15 carry K={0,1}, lanes 16-31 carry K={2,3}.
        const float ax = hi ? p[4 * k + 2] : p[4 * k + 0];
        const float ay = hi ? p[4 * k + 3] : p[4 * k + 1];
        v2f a; a.x = ax * inv; a.y = ay * inv;
        v2f b; b.x = bl[2 * k + 0]; b.y = bl[2 * k + 1];
        acc = __builtin_amdgcn_wmma_f32_16x16x4_f32(
            /*neg_a=*/false, a, /*neg_b=*/false, b,
            /*c_mod=*/(short)0, acc, /*reuse_a=*/false, /*reuse_b=*/false);
    }

    // D 16x16: VGPR r -> rows r (lanes 0-15) and r+8 (lanes 16-31), N = lane&15.
    const int j = lane & 15;             // coefficient index
    if (j < 4) {
        const int mbase = hi ? 8 : 0;
        #pragma unroll
        for (int r = 0; r < 8; ++r) {
            const int gn = tile * 16 + mbase + r;
            if (gn < TOTAL_NODES) coeff[gn * 4 + j] = acc[r];
        }
    }
}

// ---------------------------------------------------------------------------
// Kernel 2: 128-cell tile per workgroup, double-buffered activations in LDS.
// Each thread owns 2 consecutive cells -> b64 LDS loads/stores. All lanes of
// a wave share one node -> readfirstlane makes coeff/idx fetches scalar.
// Gate eval: c0 + c1*a + c2*b + c3*a*b  ==  fma(a, fma(c3,b,c1), fma(c2,b,c0)).
// ---------------------------------------------------------------------------
__global__ __launch_bounds__(THREADS)
void logicgate_main_kernel(const float* __restrict__ x,
                           const int*   __restrict__ idxA,
                           const int*   __restrict__ idxB,
                           const float* __restrict__ coeff,
                           float*       __restrict__ out)
{
    __shared__ float buf[2][128 * LSTRIDE];   // 132 KB -> 2 workgroups / WGP

    const int tid      = threadIdx.x;
    const int cellBase = blockIdx.x * CELLS;

    // Stage x: (cell,16) row-major -> LDS [node][cell]. float4 global reads
    // (coalesced b128); 4 b32 LDS stores each, 32 distinct banks per wave.
    const float4* __restrict__ xv = (const float4*)x;
    for (int i = tid; i < 16 * CELLS / 4; i += THREADS) {   // 512 float4's
        const int cell = i >> 2;
        const int ng   = (i & 3) * 4;
        const float4 t = xv[(cellBase + cell) * 4 + (i & 3)];
        buf[0][(ng + 0) * LSTRIDE + cell] = t.x;
        buf[0][(ng + 1) * LSTRIDE + cell] = t.y;
        buf[0][(ng + 2) * LSTRIDE + cell] = t.z;
        buf[0][(ng + 3) * LSTRIDE + cell] = t.w;
    }
    __syncthreads();

    const int cp      = (tid & 63) * 2;   // this thread's first cell (pair: cp, cp+1)
    const int quarter = tid >> 6;         // 0..3: node residue this thread covers
    const float4* __restrict__ cc4 = (const float4*)coeff;

    int cur = 0;   // buffer holding current layer input
    int s   = 0;   // node offset of current layer
    for (int layer = 0; layer < N_LAYERS; ++layer) {
        const int n = (layer < 16) ? 128 : (128 >> (layer - 15));
        const float* __restrict__ in = buf[cur];
        float* __restrict__ ol       = buf[cur ^ 1];

        for (int node = quarter; node < n; node += 4) {
            const int g = __builtin_amdgcn_readfirstlane(s + node);  // wave-uniform
            const float4 c = cc4[g];                                  // one b128 fetch
            const int ia = idxA[g];
            const int ib = idxB[g];
            const float2 a = *(const float2*)&in[ia * LSTRIDE + cp];  // ds_load_b64
            const float2 b = *(const float2*)&in[ib * LSTRIDE + cp];  // ds_load_b64
            float2 r;
            r.x = fmaf(a.x, fmaf(c.w, b.x, c.y), fmaf(c.z, b.x, c.x));
            r.y = fmaf(a.y, fmaf(c.w, b.y, c.y), fmaf(c.z, b.y, c.x));
            *(float2*)&ol[node * LSTRIDE + cp] = r;                   // ds_store_b64
        }
        __syncthreads();
        cur ^= 1;
        s   += n;
    }

    // Last layer width is 1: row 0 of buf[cur].
    if (tid < CELLS) out[cellBase + tid] = buf[cur][tid];
}

// ---------------------------------------------------------------------------
extern "C" void kernel_launch(void* const* d_in, const int* in_sizes, int n_in,
                              void* d_out, int out_size, void* d_ws, size_t ws_size,
                              hipStream_t stream)
{
    const float* x     = (const float*)d_in[0];   // (262144, 16)
    const float* W     = (const float*)d_in[1];   // (2175, 16)
    const int*   idx_a = (const int*)  d_in[2];   // (2175,)
    const int*   idx_b = (const int*)  d_in[3];   // (2175,)
    float*       o     = (float*)d_out;           // (262144,)
    float*       coeff = (float*)d_ws;            // 2175*4 floats scratch

    const int nTiles = (TOTAL_NODES + 15) / 16;   // 136
    logicgate_coeff_kernel<<<nTiles, 32, 0, stream>>>(W, coeff);

    const int nBlocks = N_CELLS / CELLS;          // 2048
    logicgate_main_kernel<<<nBlocks, THREADS, 0, stream>>>(x, idx_a, idx_b, coeff, o);
}